// RNN_man_76888504533718
// MI455X (gfx1250) — compile-verified
//
#include <hip/hip_runtime.h>

// Multi-layer Elman RNN (relu), fp32, T=2048 B=128 I=H=64 L=4.
// Single persistent 1024-thread workgroup (32 wave32s) on one WGP.
// Cross-layer software pipeline: while layer l is finished (inp-GEMM + relu +
// store), the NEXT layer's recurrent GEMM (h[l+1] @ W_hh^T, independent of
// layer l's output) is pre-computed into fresh accumulators -> one barrier per
// layer instead of two, and 16 extra independent WMMAs overlap the publish.
// fp32 WMMA v_wmma_f32_16x16x4_f32; fragments preloaded into register arrays.
// x[t+1] staged via GLOBAL_LOAD_ASYNC_TO_LDS_B128 (ASYNCcnt) when available.

#define T_STEPS 2048
#define BATCH   128
#define HID     64
#define LAYERS  4
#define HSTR    68   // h/x LDS row stride: A-frag b64 loads conflict-free, 16B-aligned rows
#define WSTR    72   // weight LDS row stride: B-frag b32 halves hit disjoint bank ranges

#define ENABLE_ASYNC_XSTAGE 1
#if ENABLE_ASYNC_XSTAGE && defined(__has_builtin)
#if __has_builtin(__builtin_amdgcn_global_load_async_to_lds_b128)
#define USE_ASYNC 1
#endif
#endif

typedef __attribute__((ext_vector_type(2))) float v2f;
typedef __attribute__((ext_vector_type(8))) float v8f;
typedef __attribute__((ext_vector_type(4))) int   i4;   // async builtin pointee type

// One 16x16x64 GEMM: 16 fragment pairs preloaded, then 16 WMMAs on 2 chains.
static __device__ __forceinline__ void gemm16(const float* __restrict__ amat,
                                              const float* __restrict__ wmat,
                                              int arow_off, int bcol, int brow,
                                              v8f& c0, v8f& c1) {
  v2f af[16], bf[16];
  #pragma unroll
  for (int kk = 0; kk < 16; ++kk) {
    const int k = kk * 4;
    af[kk] = *(const v2f*)(amat + arow_off + k);
    bf[kk].x = wmat[(k + brow    ) * WSTR + bcol];
    bf[kk].y = wmat[(k + brow + 1) * WSTR + bcol];
  }
  #pragma unroll
  for (int kk = 0; kk < 16; kk += 2) {
    c0 = __builtin_amdgcn_wmma_f32_16x16x4_f32(false, af[kk],     false, bf[kk],
                                               (short)0, c0, false, false);
    c1 = __builtin_amdgcn_wmma_f32_16x16x4_f32(false, af[kk + 1], false, bf[kk + 1],
                                               (short)0, c1, false, false);
  }
}

__launch_bounds__(1024, 1)
__global__ void rnn_relu_wmma_kernel(const float* __restrict__ x,
                                     const float* __restrict__ w_ih,
                                     const float* __restrict__ w_hh,
                                     const float* __restrict__ b_ih,
                                     const float* __restrict__ b_hh,
                                     float* __restrict__ out) {
  // Transposed weights: sW[(l*2+m)*HID + k][j] = W_m[l][j][k]  (m: 0=ih, 1=hh)
  __shared__ float sW[2 * LAYERS * HID * WSTR];    // 147,456 B
  __shared__ float sH[LAYERS * BATCH * HSTR];      // 139,264 B
  __shared__ float sX[BATCH * HSTR];               //  34,816 B
  __shared__ float sB[LAYERS * HID];               //   1,024 B

  const int tid  = threadIdx.x;
  const int lane = tid & 31;
  const int wave = tid >> 5;
  const int half = lane >> 4;        // 0: lanes 0-15, 1: lanes 16-31
  const int l16  = lane & 15;
  const int m0   = (wave >> 2) * 16; // batch-tile origin (8 tiles)
  const int n0   = (wave & 3) * 16;  // hidden-tile origin (4 tiles)

  // ---- one-time setup: transpose weights into LDS, fuse biases, zero h ----
  for (int i = tid; i < LAYERS * HID * HID; i += 1024) {
    int l = i >> 12;
    int r = i & 4095;
    int j = r >> 6;            // output unit (row of W)
    int k = r & 63;            // input index (col of W)
    sW[((l * 2 + 0) * HID + k) * WSTR + j] = w_ih[i];
    sW[((l * 2 + 1) * HID + k) * WSTR + j] = w_hh[i];
  }
  for (int i = tid; i < LAYERS * HID; i += 1024) sB[i] = b_ih[i] + b_hh[i];
  for (int i = tid; i < LAYERS * BATCH * HSTR; i += 1024) sH[i] = 0.0f;
  {  // stage x[0]
    const float4* xs = (const float4*)x;
    for (int f = tid; f < BATCH * HID / 4; f += 1024) {
      float4 v = xs[f];
      *(float4*)&sX[(f >> 4) * HSTR + ((f & 15) << 2)] = v;
    }
  }
  __syncthreads();

  const int arow_off = (m0 + l16) * HSTR + half * 2;  // A-fragment base offset
  const int bcol     = n0 + l16;                       // B-fragment column
  const int brow     = half * 2;                       // B-fragment first row in k-group

  // ---- prologue: recurrent GEMM for (t=0, layer 0): bias0 + h0(=0) @ W_hh0 ----
  v8f a0, a1;
  {
    float bias = sB[bcol];
    #pragma unroll
    for (int v = 0; v < 8; ++v) { a0[v] = bias; a1[v] = 0.0f; }
    gemm16(sH, &sW[1 * HID * WSTR], arow_off, bcol, brow, a0, a1);
  }
  __syncthreads();   // pre-compute reads of sH[0] done before first write

  for (int t = 0; t < T_STEPS; ++t) {
    int tn = (t + 1 < T_STEPS) ? (t + 1) : (T_STEPS - 1);
    const float* xnext = x + (size_t)tn * BATCH * HID;
#ifndef USE_ASYNC
    float4 pf0 = ((const float4*)xnext)[tid];
    float4 pf1 = ((const float4*)xnext)[tid + 1024];
#endif

    for (int l = 0; l < LAYERS; ++l) {
      if (l == 1) {
        // layer-0 barrier passed -> every wave is done reading sX: stage x[t+1]
        const int f0 = tid, f1 = tid + 1024;
#ifdef USE_ASYNC
        __builtin_amdgcn_global_load_async_to_lds_b128(
            (i4*)(xnext + f0 * 4), (i4*)&sX[(f0 >> 4) * HSTR + ((f0 & 15) << 2)], 0, 0);
        __builtin_amdgcn_global_load_async_to_lds_b128(
            (i4*)(xnext + f1 * 4), (i4*)&sX[(f1 >> 4) * HSTR + ((f1 & 15) << 2)], 0, 0);
#else
        *(float4*)&sX[(f0 >> 4) * HSTR + ((f0 & 15) << 2)] = pf0;
        *(float4*)&sX[(f1 >> 4) * HSTR + ((f1 & 15) << 2)] = pf1;
#endif
      }

      // finish layer l: accumulators already hold bias + h[l] @ W_hh^T
      const float* inp = (l == 0) ? sX : &sH[(l - 1) * BATCH * HSTR];
      gemm16(inp, &sW[(l * 2 + 0) * HID * WSTR], arow_off, bcol, brow, a0, a1);

      // pre-compute next layer's recurrent GEMM (independent of this layer's out)
      const int nl = (l + 1) & (LAYERS - 1);
      v8f d0, d1;
      {
        float nbias = sB[nl * HID + bcol];
        #pragma unroll
        for (int v = 0; v < 8; ++v) { d0[v] = nbias; d1[v] = 0.0f; }
        gemm16(&sH[nl * BATCH * HSTR], &sW[(nl * 2 + 1) * HID * WSTR],
               arow_off, bcol, brow, d0, d1);
      }

      v8f c;
      #pragma unroll
      for (int v = 0; v < 8; ++v) c[v] = fmaxf(a0[v] + a1[v], 0.0f);

      float* hw = &sH[l * BATCH * HSTR];
      #pragma unroll
      for (int v = 0; v < 8; ++v)
        hw[(m0 + half * 8 + v) * HSTR + n0 + l16] = c[v];

      if (l == LAYERS - 1) {
        float* o = out + (size_t)t * BATCH * HID;
        #pragma unroll
        for (int v = 0; v < 8; ++v)
          o[(m0 + half * 8 + v) * HID + n0 + l16] = c[v];
#ifdef USE_ASYNC
#if defined(__has_builtin) && __has_builtin(__builtin_amdgcn_s_wait_asynccnt)
        __builtin_amdgcn_s_wait_asynccnt(0);
#else
        asm volatile("s_wait_asynccnt 0" ::: "memory");
#endif
#endif
      }

      a0 = d0; a1 = d1;   // carry next layer's recurrent partials
      __syncthreads();    // publish h[l] / staged x; protect old-state reads
    }
  }

  // h_final: [L, B, H] appended after h_out
  for (int i = tid; i < LAYERS * BATCH * HID; i += 1024) {
    int l = i >> 13;
    int r = (i >> 6) & 127;
    int cc = i & 63;
    out[(size_t)T_STEPS * BATCH * HID + i] = sH[(l * BATCH + r) * HSTR + cc];
  }
}

extern "C" void kernel_launch(void* const* d_in, const int* in_sizes, int n_in,
                              void* d_out, int out_size, void* d_ws, size_t ws_size,
                              hipStream_t stream) {
  const float* x    = (const float*)d_in[0];
  const float* w_ih = (const float*)d_in[1];
  const float* w_hh = (const float*)d_in[2];
  const float* b_ih = (const float*)d_in[3];
  const float* b_hh = (const float*)d_in[4];
  float* out = (float*)d_out;
  (void)in_sizes; (void)n_in; (void)out_size; (void)d_ws; (void)ws_size;

  rnn_relu_wmma_kernel<<<dim3(1), dim3(1024), 0, stream>>>(x, w_ih, w_hh, b_ih, b_hh, out);
}